// PSTConv_39737037422959
// MI455X (gfx1250) — compile-verified
//
#include <hip/hip_runtime.h>
#include <hip/hip_bf16.h>
#include <cstddef>

// ---------------------------------------------------------------------------
// PSTConv for MI455X (gfx1250). fp32 WMMA (v_wmma_f32_16x16x4_f32) for both
// GEMMs; gather/max/FPS as wave32 VALU kernels.
//
// Shapes: B=8 L=6 N=2048 C=64 M=1024 MID=128 OUT=256 T=6 K=32 R^2=0.25
// ---------------------------------------------------------------------------

typedef __attribute__((ext_vector_type(2))) float v2f;
typedef __attribute__((ext_vector_type(8))) float v8f;

#define BB   8
#define LL   6
#define NN   2048
#define CC   64
#define MM   1024
#define MIDC 128
#define SFC  384     // 3*MID
#define OUTC 256
#define TTI  6
#define KK   32
#define R2   0.25f
#define NEG_INF (-3.402823466e38f)

// ---------------------------------------------------------------------------
// Kernel 1: farthest point sampling. One block per (b,t). Sequential in m,
// parallel over the 2048 candidate points. Tie-break = smallest index
// (matches jnp.argmax first-occurrence).
// Writes anchor coords directly into the new_xyzs output region.
// ---------------------------------------------------------------------------
__global__ __launch_bounds__(256) void fps_kernel(const float* __restrict__ xyzs,
                                                  float* __restrict__ out_xyz)
{
    __shared__ float dists[NN];
    __shared__ float rv[256];
    __shared__ int   ri[256];
    __shared__ float cur[3];
    __shared__ int   curfar;

    const int tid = threadIdx.x;
    const int bt  = blockIdx.x;                     // b*6 + tt (frame l == tt)
    const float* xyz = xyzs + (size_t)bt * NN * 3;
    float* anc = out_xyz + (size_t)bt * MM * 3;

    for (int p = tid; p < NN; p += 256) dists[p] = 1e10f;
    if (tid == 0) curfar = 0;
    __syncthreads();

    for (int m = 0; m < MM; ++m) {
        if (tid == 0) {
            int far = curfar;
            float x = xyz[far*3+0], y = xyz[far*3+1], z = xyz[far*3+2];
            cur[0] = x; cur[1] = y; cur[2] = z;
            anc[m*3+0] = x; anc[m*3+1] = y; anc[m*3+2] = z;
        }
        __syncthreads();
        const float cx = cur[0], cy = cur[1], cz = cur[2];

        float bv = NEG_INF; int bi = 0;
        for (int p = tid; p < NN; p += 256) {
            float dx = xyz[p*3+0]-cx, dy = xyz[p*3+1]-cy, dz = xyz[p*3+2]-cz;
            float d  = dx*dx + dy*dy + dz*dz;
            float dd = fminf(dists[p], d);
            dists[p] = dd;
            if (dd > bv) { bv = dd; bi = p; }   // ascending p -> keeps smallest idx on tie
        }
        rv[tid] = bv; ri[tid] = bi;
        __syncthreads();
        for (int s = 128; s > 0; s >>= 1) {
            if (tid < s) {
                float ov = rv[tid+s]; int oi = ri[tid+s];
                if (ov > rv[tid] || (ov == rv[tid] && oi < ri[tid])) { rv[tid] = ov; ri[tid] = oi; }
            }
            __syncthreads();
        }
        if (tid == 0) curfar = ri[0];
        __syncthreads();
    }
}

// ---------------------------------------------------------------------------
// Kernel 2: per-frame precompute  G[f][n][o] = sum_c w_f[o][c]*feat[f][c][n]
//                                            + sum_c w_d[o][c]*xyz[f][n][c]
// One wave per 16-row n-tile; 8 accumulators cover all 128 output channels.
// K padded 67 -> 68 (17 steps of 4). fp32 WMMA.
// A tile (16n x 4c): lane<16 holds K=r in vgpr r; lane>=16 holds K=r+2.
// B tile (4c x 16o): mirrored. D: vgpr r = rows r / r+8, col = lane&15.
// ---------------------------------------------------------------------------
__global__ __launch_bounds__(256) void g_kernel(const float* __restrict__ xyzs,
                                                const float* __restrict__ feats,
                                                const float* __restrict__ w_d,
                                                const float* __restrict__ w_f,
                                                float* __restrict__ G)
{
    const int lane = threadIdx.x & 31;
    const int wav  = blockIdx.x * 8 + (threadIdx.x >> 5);   // [0, 48*128)
    const int f    = wav >> 7;
    const int n0   = (wav & 127) << 4;
    const int lrow = lane & 15;
    const int lhi  = lane >> 4;

    const float* feat = feats + (size_t)f * CC * NN;
    const float* xyz  = xyzs  + (size_t)f * NN * 3;
    const int n = n0 + lrow;

    v8f acc[8] = {};
    for (int c0 = 0; c0 <= 64; c0 += 4) {
        v2f a;
#pragma unroll
        for (int r = 0; r < 2; ++r) {
            int c = c0 + lhi*2 + r;
            float v;
            if (c < 64)      v = feat[(size_t)c * NN + n];
            else if (c < 67) v = xyz[(size_t)n * 3 + (c - 64)];
            else             v = 0.0f;
            a[r] = v;
        }
#pragma unroll
        for (int ot = 0; ot < 8; ++ot) {
            int o = ot*16 + lrow;
            v2f bm;
#pragma unroll
            for (int r = 0; r < 2; ++r) {
                int c = c0 + lhi*2 + r;
                float v;
                if (c < 64)      v = w_f[o*64 + c];
                else if (c < 67) v = w_d[o*3 + (c - 64)];
                else             v = 0.0f;
                bm[r] = v;
            }
            acc[ot] = __builtin_amdgcn_wmma_f32_16x16x4_f32(
                false, a, false, bm, (short)0, acc[ot], false, false);
        }
    }

    float* Gf = G + (size_t)f * NN * MIDC;
#pragma unroll
    for (int ot = 0; ot < 8; ++ot)
#pragma unroll
        for (int vr = 0; vr < 8; ++vr) {
            int nr = n0 + vr + lhi*8;
            Gf[(size_t)nr * MIDC + ot*16 + lrow] = acc[ot][vr];
        }
}

// ---------------------------------------------------------------------------
// Kernel 3: ball query. One thread per anchor; block covers 256 anchors of a
// single (b,tt,dt) so frame points are staged through LDS in tiles.
// Takes the first 32 indices (index order == K smallest, exactly the
// reference top_k semantics); remainder filled with first hit (or 0).
// ---------------------------------------------------------------------------
__global__ __launch_bounds__(256) void bq_kernel(const float* __restrict__ xyzs,
                                                 const float* __restrict__ anchors,
                                                 int* __restrict__ idxbuf)
{
    __shared__ float px[256 * 3];
    const int tid  = threadIdx.x;
    const int gb   = blockIdx.x;          // ((b*6+tt)*3 + dt)*4 + mblk
    const int mblk = gb & 3;
    const int g2   = gb >> 2;
    const int dt   = g2 % 3;
    const int bt   = g2 / 3;              // b*6+tt
    const int tt   = bt % 6;
    const int l    = tt + dt - 1;
    if (l < 0 || l >= LL) return;         // padded frame: uniform over block

    const int f = (bt / 6) * 6 + l;
    const int m = mblk * 256 + tid;
    const float* xyz = xyzs    + (size_t)f  * NN * 3;
    const float* anc = anchors + (size_t)bt * MM * 3;
    int* out = idxbuf + ((size_t)(bt*3 + dt) * MM + m) * KK;

    const float ax = anc[m*3+0], ay = anc[m*3+1], az = anc[m*3+2];
    int cnt = 0, first = 0;

    for (int t0 = 0; t0 < NN; t0 += 256) {
        __syncthreads();
        int pn = t0 + tid;
        px[tid*3+0] = xyz[pn*3+0];
        px[tid*3+1] = xyz[pn*3+1];
        px[tid*3+2] = xyz[pn*3+2];
        __syncthreads();
        if (cnt < KK) {
            for (int j = 0; j < 256 && cnt < KK; ++j) {
                float dx = px[j*3+0]-ax, dy = px[j*3+1]-ay, dz = px[j*3+2]-az;
                if (dx*dx + dy*dy + dz*dz < R2) {
                    if (cnt == 0) first = t0 + j;
                    out[cnt++] = t0 + j;
                }
            }
        }
    }
    const int fill = (cnt > 0) ? first : 0;
    for (int k = cnt; k < KK; ++k) out[k] = fill;
}

// ---------------------------------------------------------------------------
// Kernel 4: gather-max + anchor correction + relu.
// One wave per (b,tt,dt,m). Lane owns 4 of 128 channels (float4, coalesced
// 512B rows of G). Index broadcast via __shfl. Padded frame => contribution
// is exactly relu(-w_d . anchor) (feat=0, g_xyz=0): use mx=0.
// Writes S[b][tt][m][dt*128 + o] (relu'd, ready for w_t GEMM).
// ---------------------------------------------------------------------------
__global__ __launch_bounds__(256) void gmax_kernel(const float* __restrict__ G,
                                                   const float* __restrict__ anchors,
                                                   const float* __restrict__ w_d,
                                                   const int* __restrict__ idxbuf,
                                                   float* __restrict__ S)
{
    const int lane = threadIdx.x & 31;
    const int wav  = blockIdx.x * 8 + (threadIdx.x >> 5);   // [0, 48*3*1024)
    const int m    = wav & (MM - 1);
    const int rest = wav >> 10;
    const int dt   = rest % 3;
    const int bt   = rest / 3;            // b*6+tt
    const int tt   = bt % 6;
    const int l    = tt + dt - 1;

    float mx0 = 0.f, mx1 = 0.f, mx2 = 0.f, mx3 = 0.f;
    if (l >= 0 && l < LL) {
        const int f = (bt / 6) * 6 + l;
        const float* Gf = G + (size_t)f * NN * MIDC;
        const int myidx = idxbuf[((size_t)(bt*3 + dt) * MM + m) * KK + lane];
        mx0 = mx1 = mx2 = mx3 = NEG_INF;
        for (int k = 0; k < KK; ++k) {
            int n = __shfl(myidx, k, 32);
            const float4 g = reinterpret_cast<const float4*>(Gf + (size_t)n * MIDC)[lane];
            mx0 = fmaxf(mx0, g.x); mx1 = fmaxf(mx1, g.y);
            mx2 = fmaxf(mx2, g.z); mx3 = fmaxf(mx3, g.w);
        }
    }

    const float ax = anchors[((size_t)bt*MM + m)*3 + 0];
    const float ay = anchors[((size_t)bt*MM + m)*3 + 1];
    const float az = anchors[((size_t)bt*MM + m)*3 + 2];
    const int o = lane * 4;
    float4 s;
    {
        float a0 = w_d[(o+0)*3+0]*ax + w_d[(o+0)*3+1]*ay + w_d[(o+0)*3+2]*az;
        float a1 = w_d[(o+1)*3+0]*ax + w_d[(o+1)*3+1]*ay + w_d[(o+1)*3+2]*az;
        float a2 = w_d[(o+2)*3+0]*ax + w_d[(o+2)*3+1]*ay + w_d[(o+2)*3+2]*az;
        float a3 = w_d[(o+3)*3+0]*ax + w_d[(o+3)*3+1]*ay + w_d[(o+3)*3+2]*az;
        s.x = fmaxf(mx0 - a0, 0.f); s.y = fmaxf(mx1 - a1, 0.f);
        s.z = fmaxf(mx2 - a2, 0.f); s.w = fmaxf(mx3 - a3, 0.f);
    }
    float* Sp = S + ((size_t)bt * MM + m) * SFC + dt * MIDC;
    reinterpret_cast<float4*>(Sp)[lane] = s;
}

// ---------------------------------------------------------------------------
// Kernel 5: final GEMM  Y[o2][m] = sum_j w_t[o2][j] * S[m][j]   (384-deep)
// per (b,tt). fp32 WMMA; wave = one 16-wide m-tile x 8 o2-tiles (half of 256).
// ---------------------------------------------------------------------------
__global__ __launch_bounds__(256) void out_kernel(const float* __restrict__ S,
                                                  const float* __restrict__ w_t,
                                                  float* __restrict__ outF)
{
    const int lane = threadIdx.x & 31;
    const int wav  = blockIdx.x * 8 + (threadIdx.x >> 5);   // [0, 48*64*2)
    const int half = wav & 1;
    const int mt   = (wav >> 1) & 63;
    const int bt   = wav >> 7;
    const int m0   = mt * 16;
    const int lrow = lane & 15;
    const int lhi  = lane >> 4;

    const float* Sb = S + (size_t)bt * MM * SFC;
    v8f acc[8] = {};

    for (int j0 = 0; j0 < SFC; j0 += 4) {
        const int jc = j0 + lhi * 2;
        const float* srow = Sb + (size_t)(m0 + lrow) * SFC + jc;
        v2f bm; bm[0] = srow[0]; bm[1] = srow[1];
#pragma unroll
        for (int ot = 0; ot < 8; ++ot) {
            const int o2 = half*128 + ot*16 + lrow;
            const float* wrow = w_t + (size_t)o2 * SFC + jc;
            v2f a; a[0] = wrow[0]; a[1] = wrow[1];
            acc[ot] = __builtin_amdgcn_wmma_f32_16x16x4_f32(
                false, a, false, bm, (short)0, acc[ot], false, false);
        }
    }

    float* Y = outF + (size_t)bt * OUTC * MM;
#pragma unroll
    for (int ot = 0; ot < 8; ++ot)
#pragma unroll
        for (int vr = 0; vr < 8; ++vr) {
            int orow = half*128 + ot*16 + vr + lhi*8;
            Y[(size_t)orow * MM + m0 + lrow] = acc[ot][vr];
        }
}

// ---------------------------------------------------------------------------
extern "C" void kernel_launch(void* const* d_in, const int* in_sizes, int n_in,
                              void* d_out, int out_size, void* d_ws, size_t ws_size,
                              hipStream_t stream) {
    (void)in_sizes; (void)n_in; (void)out_size; (void)ws_size;
    const float* xyzs  = (const float*)d_in[0];   // (8,6,2048,3)
    const float* feats = (const float*)d_in[1];   // (8,6,64,2048)
    const float* w_d   = (const float*)d_in[2];   // (128,3)
    const float* w_f   = (const float*)d_in[3];   // (128,64)
    const float* w_t   = (const float*)d_in[4];   // (256,384)

    float* out_xyz  = (float*)d_out;                          // (8,6,1024,3)
    float* out_feat = out_xyz + (size_t)BB*TTI*MM*3;          // (8,6,256,1024)

    float* G   = (float*)d_ws;                                // 48*2048*128 f32
    float* S   = G + (size_t)BB*LL*NN*MIDC;                   // 48*1024*384 f32
    int*   idx = (int*)(S + (size_t)BB*TTI*MM*SFC);           // 48*3*1024*32 i32

    fps_kernel <<<BB*TTI,            256, 0, stream>>>(xyzs, out_xyz);
    g_kernel   <<<(BB*LL*(NN/16))/8, 256, 0, stream>>>(xyzs, feats, w_d, w_f, G);
    bq_kernel  <<<BB*TTI*3*(MM/256), 256, 0, stream>>>(xyzs, out_xyz, idx);
    gmax_kernel<<<(BB*TTI*3*MM)/8,   256, 0, stream>>>(G, out_xyz, w_d, idx, S);
    out_kernel <<<(BB*TTI*64*2)/8,   256, 0, stream>>>(S, w_t, out_feat);
}